// NonDiffCorrespondenceEngine_29703993819772
// MI455X (gfx1250) — compile-verified
//
#include <hip/hip_runtime.h>
#include <stdint.h>

typedef __attribute__((ext_vector_type(2))) float v2f;
typedef __attribute__((ext_vector_type(8))) float v8f;

#define NEG_INF (-__builtin_inff())

// Problem geometry
#define H0 384
#define W0 384
#define H1 192
#define W1 192
#define N0 (H0 * W0)        // 147456
#define N1 (H1 * W1)        // 36864
#define NTOT (N0 + N1)      // 184320
#define CDESC 128
#define KKP 512             // top-k keypoints
#define KM 128              // top-k matches
#define NM (KKP * KKP)      // 262144
#define SCORE_THRESH 0.7f

// gfx1250 async global->LDS path (guarded: fall back to plain copies if absent)
#if defined(__has_builtin)
#  if __has_builtin(__builtin_amdgcn_global_load_async_to_lds_b128) && \
      __has_builtin(__builtin_amdgcn_s_wait_asynccnt)
#    define USE_ASYNC_LDS 1
#  endif
#endif

#if USE_ASYNC_LDS
// Builtin signature (from hipcc diagnostic): (v4i AS1*, v4i AS3*, imm int, imm int)
typedef int async_v4i __attribute__((vector_size(16)));
typedef __attribute__((address_space(1))) async_v4i* gbl_v4i_p;
typedef __attribute__((address_space(3))) async_v4i* lds_v4i_p;
#endif

// ---------------- helpers ----------------
__device__ __forceinline__ unsigned fkey(float x) {
  unsigned u = __float_as_uint(x);
  return (u & 0x80000000u) ? ~u : (u | 0x80000000u);  // monotone: bigger float -> bigger key
}

// ---------------- init (reset selection state every call) ----------------
__global__ void init_k(float* c1s, int* c1i, float* c2s, int* c2i,
                       unsigned* st1, unsigned* st2) {
  int i = blockIdx.x * blockDim.x + threadIdx.x;
  if (i < 4 * KKP) { c1s[i] = NEG_INF; c1i[i] = 0; }
  if (i < 2 * KM)  { c2s[i] = NEG_INF; c2i[i] = 0; }
  if (i < 4 * 8)   st1[i] = ((i & 7) == 1) ? (unsigned)KKP : 0u;
  if (i < 2 * 8)   st2[i] = ((i & 7) == 1) ? (unsigned)KM  : 0u;
}

// ---------------- stage A: masked NMS score maps ----------------
// rows r = im*2 + bi  (im: image 0/1, bi: batch 0/1). S: [4][NTOT]
__global__ void score_k(const float* __restrict__ repA0, const float* __restrict__ relA0,
                        const float* __restrict__ repA1, const float* __restrict__ relA1,
                        const float* __restrict__ repB0, const float* __restrict__ relB0,
                        const float* __restrict__ repB1, const float* __restrict__ relB1,
                        float* __restrict__ S) {
  int i = blockIdx.x * 256 + threadIdx.x;
  if (i >= NTOT) return;
  int r = blockIdx.y;
  int im = r >> 1, bi = r & 1;
  const float* rep; const float* rel;
  int Ww, Hh, pix, base;
  if (i < N0) {
    rep = im ? repB0 : repA0; rel = im ? relB0 : relA0;
    Ww = W0; Hh = H0; pix = i; base = bi * N0;
  } else {
    rep = im ? repB1 : repA1; rel = im ? relB1 : relA1;
    Ww = W1; Hh = H1; pix = i - N0; base = bi * N1;
  }
  int y = pix / Ww, x = pix - y * Ww;
  float c = rep[base + pix];
  float m = c;
  #pragma unroll
  for (int dy = -1; dy <= 1; ++dy) {
    #pragma unroll
    for (int dx = -1; dx <= 1; ++dx) {
      int yy = y + dy, xx = x + dx;
      if (yy >= 0 && yy < Hh && xx >= 0 && xx < Ww)
        m = fmaxf(m, rep[base + yy * Ww + xx]);
    }
  }
  float mean = sqrtf(c * rel[base + pix]);
  float s = ((c == m) && (mean >= SCORE_THRESH)) ? mean : NEG_INF;
  S[(size_t)r * NTOT + i] = s;
}

// ---------------- radix-select (one 8-bit digit per launch) ----------------
// state per row: [0]=prefix/threshold key, [1]=k_remaining, [2]=numHigh, [3]=ctrHigh, [4]=ctrEq
__global__ void radix_pass_k(const float* __restrict__ data, unsigned* state,
                             int N, int K, int pass) {
  int row = blockIdx.x;
  unsigned* st = state + row * 8;
  __shared__ unsigned hist[256];
  hist[threadIdx.x] = 0;
  __syncthreads();
  unsigned prefix = st[0];
  unsigned krem = st[1];
  int shift = 24 - 8 * pass;
  unsigned hmask = pass ? (0xFFFFFFFFu << (shift + 8)) : 0u;
  const float* d = data + (size_t)row * N;
  for (int i = threadIdx.x; i < N; i += 256) {
    unsigned key = fkey(d[i]);
    if ((key & hmask) == (prefix & hmask))
      atomicAdd(&hist[(key >> shift) & 0xFFu], 1u);
  }
  __syncthreads();
  if (threadIdx.x == 0) {
    unsigned cum = 0; int sel = 0;
    for (int b = 255; b >= 0; --b) {
      unsigned c = hist[b];
      if (cum + c >= krem) { sel = b; break; }
      cum += c;
    }
    st[0] = prefix | ((unsigned)sel << shift);
    unsigned nk = (krem > cum) ? (krem - cum) : 0u;
    st[1] = nk;
    if (pass == 3) st[2] = (unsigned)K - nk;  // numHigh
  }
}

__global__ void compact_k(const float* __restrict__ data, unsigned* state,
                          float* __restrict__ cs, int* __restrict__ ci, int N, int K) {
  int row = blockIdx.y;
  unsigned* st = state + row * 8;
  int i = blockIdx.x * 256 + threadIdx.x;
  if (i >= N) return;
  float x = data[(size_t)row * N + i];
  unsigned key = fkey(x);
  unsigned T = st[0], nk = st[1], nh = st[2];
  int slot = -1;
  if (key > T) {
    slot = (int)atomicAdd(&st[3], 1u);
  } else if (key == T) {
    unsigned e = atomicAdd(&st[4], 1u);
    if (e < nk) slot = (int)(nh + e);
  }
  if (slot >= 0 && slot < K) { cs[row * K + slot] = x; ci[row * K + slot] = i; }
}

// ---------------- in-LDS bitonic sort of K (power-of-2) candidates ----------------
__global__ void sort_k(float* cs, int* ci, int K) {
  extern __shared__ float sm[];
  float* ss = sm;
  int* si = (int*)(sm + K);
  int row = blockIdx.x, tid = threadIdx.x;
  ss[tid] = cs[row * K + tid];
  si[tid] = ci[row * K + tid];
  __syncthreads();
  for (int k = 2; k <= K; k <<= 1) {
    for (int j = k >> 1; j > 0; j >>= 1) {
      int ixj = tid ^ j;
      if (ixj > tid) {
        float s0 = ss[tid], s1 = ss[ixj];
        int i0 = si[tid], i1 = si[ixj];
        bool before = (s0 > s1) || (s0 == s1 && i0 < i1);  // descending, ties by index
        bool up = ((tid & k) == 0);
        if (up ? !before : before) {
          ss[tid] = s1; ss[ixj] = s0; si[tid] = i1; si[ixj] = i0;
        }
      }
      __syncthreads();
    }
  }
  cs[row * K + tid] = ss[tid];
  ci[row * K + tid] = si[tid];
}

// ---------------- gather descriptors (scaled by score) + locations ----------------
// image1 -> D1 [b][k][c] (K-major, for WMMA A); image2 -> D2T [b][c][l] (C-major, for WMMA B)
__global__ void gather_k(const float* __restrict__ dA0, const float* __restrict__ dA1,
                         const float* __restrict__ dB0, const float* __restrict__ dB1,
                         const float* __restrict__ c1s, const int* __restrict__ c1i,
                         float* __restrict__ D1, float* __restrict__ D2T,
                         float* __restrict__ Loc) {
  int r = blockIdx.y;             // 0..3
  int k = blockIdx.x;             // 0..511
  int c = threadIdx.x;            // 0..127
  int im = r >> 1, bi = r & 1;
  float s = c1s[r * KKP + k];
  int pix = c1i[r * KKP + k];
  bool valid = (s != NEG_INF);
  float sf = valid ? s : 0.0f;
  float v, locy, locx;
  if (pix < N0) {
    const float* dsc = im ? dB0 : dA0;
    v = dsc[(size_t)bi * CDESC * N0 + (size_t)c * N0 + pix];
    locy = (float)(pix / W0); locx = (float)(pix % W0);
  } else {
    const float* dsc = im ? dB1 : dA1;
    int p1 = pix - N0;
    v = dsc[(size_t)bi * CDESC * N1 + (size_t)c * N1 + p1];
    locy = (float)(p1 / W1) * 2.0f; locx = (float)(p1 % W1) * 2.0f;
  }
  float val = v * sf;
  if (im == 0) D1[(size_t)bi * KKP * CDESC + (size_t)k * CDESC + c] = val;
  else         D2T[(size_t)bi * CDESC * KKP + (size_t)c * KKP + k] = val;
  if (c == 0) {
    float* L = Loc + ((size_t)r * KKP + k) * 2;
    L[0] = valid ? locy : 0.0f;
    L[1] = valid ? locx : 0.0f;
  }
}

// ---------------- WMMA GEMM: M[b][k][l] = sum_c D1[b][k][c] * D2T[b][c][l] ----------------
// 4 waves per block share one A tile (16x128) staged to LDS via gfx1250 async
// global->LDS DMA; each wave owns a 16x16 output tile and runs 32 fully-unrolled
// V_WMMA_F32_16X16X4_F32 over C=128. A pitch = 132 floats -> conflict-free ds_load_b64.
#define GEMM_WAVES 4
#define APITCH 132

__global__ __launch_bounds__(32 * GEMM_WAVES)
void wmma_gemm_k(const float* __restrict__ D1, const float* __restrict__ D2T,
                 float* __restrict__ M) {
  __shared__ __align__(16) float As[16 * APITCH];
  int tid = threadIdx.x;          // 0..127
  int lane = tid & 31;
  int w = tid >> 5;               // wave id 0..3
  int kt = blockIdx.x * 16;
  int lt = (blockIdx.y * GEMM_WAVES + w) * 16;
  int b = blockIdx.z;

  // stage A tile (16 rows x 128 c): thread t copies 16 consecutive floats of one row
  {
    int r0 = tid >> 3;            // 0..15
    int c0 = (tid & 7) * 16;      // 0,16,...,112
    const float* src = D1 + (size_t)b * KKP * CDESC + (size_t)(kt + r0) * CDESC + c0;
    float* dst = &As[r0 * APITCH + c0];
#if USE_ASYNC_LDS
    #pragma unroll
    for (int q = 0; q < 4; ++q)
      __builtin_amdgcn_global_load_async_to_lds_b128(
          (gbl_v4i_p)(float*)(src + 4 * q),
          (lds_v4i_p)(dst + 4 * q),
          /*offset=*/0, /*cpol=*/0);
    __builtin_amdgcn_s_wait_asynccnt(0);
#else
    #pragma unroll
    for (int q = 0; q < 4; ++q)
      *(float4*)(dst + 4 * q) = *(const float4*)(src + 4 * q);
#endif
  }
  __syncthreads();

  // A fragment (16x4 f32): lanes 0-15 rows M, K = c0,c0+1; lanes 16-31 rows M, K = c0+2,c0+3
  const float* Al = &As[(lane & 15) * APITCH + 2 * (lane >> 4)];
  // B fragment (4x16 f32): VGPR0 lanes0-15 -> (K=c0, N), lanes16-31 -> (K=c0+1, N)
  const float* Bm = D2T + (size_t)b * CDESC * KKP
                       + (size_t)(lane >> 4) * KKP + lt + (lane & 15);
  v8f acc = {};
  #pragma unroll
  for (int c0 = 0; c0 < CDESC; c0 += 4) {
    v2f a;  a.x = Al[c0];  a.y = Al[c0 + 1];
    v2f bb; bb.x = Bm[(size_t)c0 * KKP]; bb.y = Bm[(size_t)(c0 + 2) * KKP];
    acc = __builtin_amdgcn_wmma_f32_16x16x4_f32(
        /*neg_a=*/false, a, /*neg_b=*/false, bb,
        /*c_mod=*/(short)0, acc, /*reuse_a=*/false, /*reuse_b=*/false);
  }
  // C/D layout: VGPR r, lane L -> M = r + 8*(L/16), N = L%16
  float* Out = M + (size_t)b * NM + (size_t)(kt + 8 * (lane >> 4)) * KKP + lt + (lane & 15);
  #pragma unroll
  for (int r = 0; r < 8; ++r) Out[(size_t)r * KKP] = acc[r];
}

// ---------------- final output: (b, 4, 128) = [l1_y, l1_x, l2_y, l2_x] ----------------
__global__ void out_k(const int* __restrict__ ci2, const float* __restrict__ Loc,
                      float* __restrict__ out) {
  int b = blockIdx.x;     // 0..1
  int m = threadIdx.x;    // 0..127
  int idx = ci2[b * KM + m];
  int i1 = idx / KKP, i2 = idx % KKP;
  const float* l1 = Loc + ((size_t)(0 * 2 + b) * KKP + i1) * 2;  // image1 rows 0,1
  const float* l2 = Loc + ((size_t)(2 + b) * KKP + i2) * 2;      // image2 rows 2,3
  float* o = out + (size_t)b * 4 * KM + m;
  o[0 * KM] = l1[0];
  o[1 * KM] = l1[1];
  o[2 * KM] = l2[0];
  o[3 * KM] = l2[1];
}

// ---------------- launch ----------------
extern "C" void kernel_launch(void* const* d_in, const int* in_sizes, int n_in,
                              void* d_out, int out_size, void* d_ws, size_t ws_size,
                              hipStream_t stream) {
  (void)in_sizes; (void)n_in; (void)out_size; (void)ws_size;
  const float* rep1_0  = (const float*)d_in[0];
  const float* rel1_0  = (const float*)d_in[1];
  const float* desc1_0 = (const float*)d_in[2];
  const float* rep1_1  = (const float*)d_in[3];
  const float* rel1_1  = (const float*)d_in[4];
  const float* desc1_1 = (const float*)d_in[5];
  const float* rep2_0  = (const float*)d_in[6];
  const float* rel2_0  = (const float*)d_in[7];
  const float* desc2_0 = (const float*)d_in[8];
  const float* rep2_1  = (const float*)d_in[9];
  const float* rel2_1  = (const float*)d_in[10];
  const float* desc2_1 = (const float*)d_in[11];
  float* out = (float*)d_out;

  // workspace layout (4-byte units)
  float* W = (float*)d_ws;
  float*    SC  = W + 0;            // 4 * 184320
  float*    M   = W + 737280;       // 2 * 512*512
  float*    D1  = W + 1261568;      // 2 * 512*128
  float*    D2T = W + 1392640;      // 2 * 128*512
  float*    LOC = W + 1523712;      // 4 * 512 * 2
  float*    C1S = W + 1527808;      // 4 * 512
  int*      C1I = (int*)(W + 1529856);
  float*    C2S = W + 1531904;      // 2 * 128
  int*      C2I = (int*)(W + 1532160);
  unsigned* ST1 = (unsigned*)(W + 1532416);  // 4 * 8
  unsigned* ST2 = (unsigned*)(W + 1532448);  // 2 * 8

  // 1) reset per-call selection state
  init_k<<<8, 256, 0, stream>>>(C1S, C1I, C2S, C2I, ST1, ST2);

  // 2) score maps (NMS + sqrt(rep*rel) threshold), 4 rows
  score_k<<<dim3(NTOT / 256, 4), 256, 0, stream>>>(
      rep1_0, rel1_0, rep1_1, rel1_1, rep2_0, rel2_0, rep2_1, rel2_1, SC);

  // 3) top-512 keypoints per row: 4-pass radix select + compact + bitonic sort
  for (int p = 0; p < 4; ++p)
    radix_pass_k<<<4, 256, 0, stream>>>(SC, ST1, NTOT, KKP, p);
  compact_k<<<dim3(NTOT / 256, 4), 256, 0, stream>>>(SC, ST1, C1S, C1I, NTOT, KKP);
  sort_k<<<4, KKP, KKP * 8, stream>>>(C1S, C1I, KKP);

  // 4) gather scaled descriptors + locations
  gather_k<<<dim3(KKP, 4), CDESC, 0, stream>>>(
      desc1_0, desc1_1, desc2_0, desc2_1, C1S, C1I, D1, D2T, LOC);

  // 5) match scores via WMMA f32 GEMM (2 x 512x512x128), A staged through LDS
  wmma_gemm_k<<<dim3(KKP / 16, KKP / 16 / GEMM_WAVES, 2), 32 * GEMM_WAVES, 0, stream>>>(
      D1, D2T, M);

  // 6) top-128 matches per batch
  for (int p = 0; p < 4; ++p)
    radix_pass_k<<<2, 256, 0, stream>>>(M, ST2, NM, KM, p);
  compact_k<<<dim3(NM / 256, 2), 256, 0, stream>>>(M, ST2, C2S, C2I, NM, KM);
  sort_k<<<2, KM, KM * 8, stream>>>(C2S, C2I, KM);

  // 7) emit matched locations (b, 4, 128)
  out_k<<<2, KM, 0, stream>>>(C2I, LOC, out);
}